// VectorQuantizer2_7524782702953
// MI455X (gfx1250) — compile-verified
//
#include <hip/hip_runtime.h>

typedef __attribute__((ext_vector_type(2))) float v2f;
typedef __attribute__((ext_vector_type(8))) float v8f;
typedef __attribute__((ext_vector_type(4))) int   v4i;

#define K_CODES   8192
#define DIM       256
#define N_ROWS    32768
#define CB_STRIDE 260   // rows are 1040B (16B aligned), conflict-free bank pattern

#if __has_builtin(__builtin_amdgcn_global_load_async_to_lds_b128) && \
    __has_builtin(__builtin_amdgcn_s_wait_asynccnt)
#define HAVE_ASYNC_LDS 1
typedef __attribute__((address_space(1))) v4i gv4i;   // global (printed __device__)
typedef __attribute__((address_space(3))) v4i lv4i;   // LDS    (printed __shared__)
#else
#define HAVE_ASYNC_LDS 0
#endif

// ---------------- kernel 0: codebook squared norms (one wave per row) ----------------
__global__ void __launch_bounds__(256) vq_norms(const float* __restrict__ cb,
                                                float* __restrict__ enorm) {
    const int wave = threadIdx.x >> 5;
    const int lane = threadIdx.x & 31;
    const int row  = blockIdx.x * 8 + wave;
    const float* r = cb + row * DIM;
    float s = 0.f;
#pragma unroll
    for (int i = 0; i < 8; ++i) { float v = r[lane + 32 * i]; s += v * v; }
#pragma unroll
    for (int off = 16; off >= 1; off >>= 1) s += __shfl_xor(s, off, 32);
    if (lane == 0) enorm[row] = s;
}

// ---------------- kernel 1: fused GEMM + argmin via V_WMMA_F32_16X16X4_F32 ----------------
__global__ void __launch_bounds__(256) vq_argmin(const float* __restrict__ z,
                                                 const float* __restrict__ cb,
                                                 const float* __restrict__ enorm,
                                                 int*   __restrict__ idx_out,
                                                 float* __restrict__ idx_out_f) {
    __shared__ float cbs[2][16 * CB_STRIDE];
    __shared__ float ens[2][16];

    const int tid   = threadIdx.x;
    const int wave  = tid >> 5;
    const int lane  = tid & 31;
    const int laneM = lane & 15;   // A: row M | B: col N | C: N
    const int laneH = lane >> 4;   // K pair select / M high bit
    const int zbase = blockIdx.x * 128;          // 8 waves * 16 rows
    const int myrow = zbase + wave * 16 + laneM;

    // --- A fragments: this wave's 16x256 z tile, VGPR-resident for all 512 chunks.
    // Layout (ISA 7.12.2): lane holds M=lane%16, K = 4t + 2*(lane/16) + {0,1}.
    v2f a[64];
    const float* zr = z + (size_t)myrow * DIM;
#pragma unroll
    for (int t = 0; t < 64; ++t)
        a[t] = *(const v2f*)(zr + 4 * t + 2 * laneH);

    float best[8];
    int   bidx[8];
#pragma unroll
    for (int j = 0; j < 8; ++j) { best[j] = 3.0e38f; bidx[j] = 0; }

    // --- staging helper: 16 codebook rows (16KB) -> LDS, 4 x b128 per thread
    auto stage = [&](int buf, int kb) {
#pragma unroll
        for (int i = 0; i < 4; ++i) {
            int li = tid + 256 * i;           // 0..1023 float4 slots
            int r  = li >> 6, c4 = li & 63;
#if HAVE_ASYNC_LDS
            __builtin_amdgcn_global_load_async_to_lds_b128(
                (gv4i*)(cb + (size_t)(kb + r) * DIM + c4 * 4),
                (lv4i*)(&cbs[buf][r * CB_STRIDE + c4 * 4]),
                0, 0);
#else
            *(float4*)&cbs[buf][r * CB_STRIDE + c4 * 4] =
                *(const float4*)&cb[(size_t)(kb + r) * DIM + c4 * 4];
#endif
        }
        if (tid < 16) ens[buf][tid] = enorm[kb + tid];
    };

    // prologue: fill buffer 0
    stage(0, 0);
#if HAVE_ASYNC_LDS
    __builtin_amdgcn_s_wait_asynccnt(0);
#endif
    __syncthreads();

    int p = 0;
    for (int kb = 0; kb < K_CODES; kb += 16) {
        // prefetch next chunk into the other buffer (hidden behind the WMMAs)
        if (kb + 16 < K_CODES) stage(p ^ 1, kb + 16);

        // 64 f32 WMMAs split into 4 independent accumulator chains (4-way ILP)
        const float* buf = cbs[p];
        v8f c0 = {}, c1 = {}, c2 = {}, c3 = {};
#pragma unroll
        for (int t = 0; t < 64; t += 4) {
            const float* bp = buf + laneM * CB_STRIDE + 2 * laneH;
            v2f b0 = *(const v2f*)(bp + 4 * (t + 0));
            v2f b1 = *(const v2f*)(bp + 4 * (t + 1));
            v2f b2 = *(const v2f*)(bp + 4 * (t + 2));
            v2f b3 = *(const v2f*)(bp + 4 * (t + 3));
            c0 = __builtin_amdgcn_wmma_f32_16x16x4_f32(false, a[t + 0], false, b0, (short)0, c0, false, false);
            c1 = __builtin_amdgcn_wmma_f32_16x16x4_f32(false, a[t + 1], false, b1, (short)0, c1, false, false);
            c2 = __builtin_amdgcn_wmma_f32_16x16x4_f32(false, a[t + 2], false, b2, (short)0, c2, false, false);
            c3 = __builtin_amdgcn_wmma_f32_16x16x4_f32(false, a[t + 3], false, b3, (short)0, c3, false, false);
        }

        // Epilogue: score = ||e||^2 - 2*dot ; per-lane running argmin.
        const float en   = ens[p][laneM];
        const int   kidx = kb + laneM;
#pragma unroll
        for (int j = 0; j < 8; ++j) {
            float s = en - 2.0f * ((c0[j] + c1[j]) + (c2[j] + c3[j]));
            if (s < best[j]) { best[j] = s; bidx[j] = kidx; }  // strict < keeps first occurrence
        }

        // publish prefetched buffer / protect current buffer from overwrite
#if HAVE_ASYNC_LDS
        __builtin_amdgcn_s_wait_asynccnt(0);
#endif
        __syncthreads();
        p ^= 1;
    }

    // Reduce over the 16 code-columns (lanes within each half-wave group),
    // lexicographic (value, index) min == jnp.argmin first-occurrence tie-break.
#pragma unroll
    for (int j = 0; j < 8; ++j) {
#pragma unroll
        for (int off = 8; off >= 1; off >>= 1) {
            float ob = __shfl_xor(best[j], off, 16);
            int   oi = __shfl_xor(bidx[j], off, 16);
            if (ob < best[j] || (ob == best[j] && oi < bidx[j])) { best[j] = ob; bidx[j] = oi; }
        }
    }
    if (laneM == 0) {
#pragma unroll
        for (int j = 0; j < 8; ++j) {      // C element M = j + 8*laneH
            int g = zbase + wave * 16 + j + 8 * laneH;
            idx_out[g]   = bidx[j];
            idx_out_f[g] = (float)bidx[j];
        }
    }
}

// ---------------- kernel 2: gather z_q + per-block partial loss ----------------
__global__ void __launch_bounds__(256) vq_gather(const float* __restrict__ z,
                                                 const float* __restrict__ cb,
                                                 const int*   __restrict__ idx,
                                                 float* __restrict__ zq_out,
                                                 float* __restrict__ partials) {
    __shared__ float red[256];
    const int r0 = blockIdx.x * 16;
    float acc = 0.f;
    for (int r = 0; r < 16; ++r) {
        const int row = r0 + r;
        const int k   = idx[row];
        float q  = cb[(size_t)k * DIM + threadIdx.x];
        float zz = z[(size_t)row * DIM + threadIdx.x];
        float d  = q - zz;
        acc += d * d;
        zq_out[(size_t)row * DIM + threadIdx.x] = q;   // z_q_st forward value == z_q
    }
    red[threadIdx.x] = acc;
    __syncthreads();
    for (int s = 128; s >= 1; s >>= 1) {
        if (threadIdx.x < s) red[threadIdx.x] += red[threadIdx.x + s];
        __syncthreads();
    }
    if (threadIdx.x == 0) partials[blockIdx.x] = red[0];
}

// ---------------- kernel 3: final deterministic loss reduction ----------------
__global__ void __launch_bounds__(256) vq_loss(const float* __restrict__ partials,
                                               float* __restrict__ loss_out) {
    __shared__ float red[256];
    float acc = 0.f;
    for (int i = threadIdx.x; i < 2048; i += 256) acc += partials[i];
    red[threadIdx.x] = acc;
    __syncthreads();
    for (int s = 128; s >= 1; s >>= 1) {
        if (threadIdx.x < s) red[threadIdx.x] += red[threadIdx.x + s];
        __syncthreads();
    }
    // legacy=False forward value: (beta + 1) * mean((z_q - z)^2)
    if (threadIdx.x == 0)
        *loss_out = 1.25f * red[0] / (float)(N_ROWS * DIM);
}

extern "C" void kernel_launch(void* const* d_in, const int* in_sizes, int n_in,
                              void* d_out, int out_size, void* d_ws, size_t ws_size,
                              hipStream_t stream) {
    const float* z  = (const float*)d_in[0];   // [8,4096,256]
    const float* cb = (const float*)d_in[1];   // [8192,256]

    float* out  = (float*)d_out;
    float* zq   = out;                              // N*D floats (z_q_st)
    float* loss = out + (size_t)N_ROWS * DIM;       // 1 float
    float* idxf = loss + 1;                         // N floats (idx)

    char*  ws       = (char*)d_ws;
    float* enorm    = (float*)ws;                                   // 8192 f32
    int*   idx      = (int*)(ws + (size_t)K_CODES * 4);             // 32768 i32
    float* partials = (float*)(ws + (size_t)K_CODES * 4 + (size_t)N_ROWS * 4); // 2048 f32

    vq_norms <<<K_CODES / 8,  256, 0, stream>>>(cb, enorm);
    vq_argmin<<<N_ROWS / 128, 256, 0, stream>>>(z, cb, enorm, idx, idxf);
    vq_gather<<<N_ROWS / 16,  256, 0, stream>>>(z, cb, idx, zq, partials);
    vq_loss  <<<1,            256, 0, stream>>>(partials, loss);
}